// DecoderWords_29360396435743
// MI455X (gfx1250) — compile-verified
//
#include <hip/hip_runtime.h>
#include <math.h>

typedef float v2f __attribute__((ext_vector_type(2)));
typedef float v8f __attribute__((ext_vector_type(8)));

#define B_  32
#define T_  128
#define H_  1024
#define V_  32000
#define G4_ 4096   // 4*H

__device__ __forceinline__ float sigm_(float x) { return 1.0f / (1.0f + expf(-x)); }

// ---------------------------------------------------------------------------
// Generic C[m][n] = sum_k Arow(m)[k] * W[n][k] + bias0[n] (+bias1[n])
//   gather=1: Arow(m) = emb[token(m)]  (token from teacher-forced labels)
//   remap =1: output row m = t*B+b is stored at row b*T+t  (for [B,T,V] logits)
// Tiling: BM=64, BN=128, BK=16; 8 waves; each wave: one 16-row M-tile,
// four 16-col N-tiles -> 4 x v8f accumulators via V_WMMA_F32_16X16X4_F32.
// LDS layouts give every WMMA fragment as ONE aligned ds_load_b64:
//   As[row][18]      : k contiguous, even row stride -> aligned v2f reads
//   BsP[kp][129][2]  : k-pairs contiguous per n      -> aligned v2f reads
// ---------------------------------------------------------------------------
__global__ __launch_bounds__(256) void gemm_aw(
    const float* __restrict__ A, int lda,
    const float* __restrict__ emb,
    const int*   __restrict__ labels,
    const float* __restrict__ W,          // [Ntot][K] row-major
    const float* __restrict__ bias0,
    const float* __restrict__ bias1,
    float* __restrict__ C, long ldc,
    int K, int gather, int remap)
{
    __shared__ float As[64][18];         // even stride -> 8B-aligned pairs
    __shared__ float BsP[8][129][2];     // [k/2][n(+pad)][k&1]

    const int tid    = threadIdx.x;
    const int m_base = blockIdx.x * 64;
    const int n_base = blockIdx.y * 128;
    const int wave = tid >> 5, lane = tid & 31;
    const int wm = wave & 3;                 // which 16-row M-tile
    const int wn = wave >> 2;                // which 64-col half of BN
    const int ml = lane & 15;
    const int khalf = (lane >> 4) << 1;      // 0 or 2 (A/B K split per ISA)
    const int khp   = khalf >> 1;            // 0 or 1 (pair index)
    const int mhalf = (lane >> 4) << 3;      // 0 or 8 (C/D M split per ISA)

    const v8f vz = {0.f,0.f,0.f,0.f,0.f,0.f,0.f,0.f};
    v8f acc[4] = { vz, vz, vz, vz };

    const int lk = tid & 15;                 // k within BK
    const int lr = tid >> 4;                 // 0..15

    for (int kk = 0; kk < K; kk += 16) {
        // stage A: 64 rows x 16 k (4 rows per thread), coalesced along k
        #pragma unroll
        for (int i = 0; i < 4; ++i) {
            int m  = lr + i * 16;
            int mg = m_base + m;
            float v;
            if (gather) {
                int t = mg >> 5, b = mg & 31;          // m = t*B + b, B=32
                int tok = (t == 0) ? 0 : labels[b * T_ + t - 1];
                v = emb[(size_t)tok * (size_t)K + kk + lk];
            } else {
                v = A[(size_t)mg * (size_t)lda + kk + lk];
            }
            As[m][lk] = v;
        }
        // stage B: BsP[k>>1][n][k&1] = W[n_base+n][kk+k], 8 n per thread
        #pragma unroll
        for (int i = 0; i < 8; ++i) {
            int n = lr + i * 16;
            BsP[lk >> 1][n][lk & 1] = W[(size_t)(n_base + n) * (size_t)K + kk + lk];
        }
        // prefetch next B tile (global_prefetch_b8) while this stage computes
        if (kk + 16 < K)
            __builtin_prefetch(W + (size_t)(n_base + lr) * (size_t)K + kk + 16 + lk, 0, 1);
        __syncthreads();

        // pre-load ALL fragments for the stage (clauseable ds_load_b64s)
        v2f af[4];
        v2f bf[4][4];
        #pragma unroll
        for (int k4 = 0; k4 < 4; ++k4) {
            int kb = k4 * 4 + khalf;
            int kp = k4 * 2 + khp;
            af[k4] = *reinterpret_cast<const v2f*>(&As[wm * 16 + ml][kb]);
            #pragma unroll
            for (int j = 0; j < 4; ++j)
                bf[k4][j] = *reinterpret_cast<const v2f*>(&BsP[kp][wn * 64 + j * 16 + ml][0]);
        }
        // dense WMMA burst
        #pragma unroll
        for (int k4 = 0; k4 < 4; ++k4)
            #pragma unroll
            for (int j = 0; j < 4; ++j)
                acc[j] = __builtin_amdgcn_wmma_f32_16x16x4_f32(
                    false, af[k4], false, bf[k4][j], (short)0, acc[j], false, false);
        __syncthreads();
    }

    #pragma unroll
    for (int j = 0; j < 4; ++j) {
        int ncol = n_base + wn * 64 + j * 16 + ml;
        float bv = 0.f;
        if (bias0) bv += bias0[ncol];
        if (bias1) bv += bias1[ncol];
        #pragma unroll
        for (int r = 0; r < 8; ++r) {
            int  m   = m_base + wm * 16 + r + mhalf;
            long row = remap ? (long)((m & 31) * T_ + (m >> 5)) : (long)m;
            C[row * ldc + ncol] = acc[j][r] + bv;
        }
    }
}

// ---------------------------------------------------------------------------
// One recurrent LSTM time step for one layer.
// grid 16 x 256: block handles n in [n0, n0+64) across all 4 gates:
//   gates[B=32][4x64] = h_prev @ W_hh[cols]^T  (WMMA), + pre (input+bias),
//   then cell update, h -> hs[t], c -> cbuf.
// 8 waves: wave = (gate<<1)|mtile; each wave has 4 WMMA accumulators.
// ---------------------------------------------------------------------------
__global__ __launch_bounds__(256) void lstm_step(
    const float* __restrict__ pre,    // [T*B][4H] input contribution (+biases)
    const float* __restrict__ Whh,    // [4H][H]
    const float* __restrict__ h0,     // [B][H] initial hidden (layer slice)
    const float* __restrict__ c0,     // [B][H] initial cell   (layer slice)
    float* __restrict__ hs,           // [T*B][H]
    float* __restrict__ cbuf,         // [B][H] running cell state
    int t)
{
    __shared__ float As[32][18];
    __shared__ float BsP[8][257][2];  // 256 gate-cols (+pad), k-pair-major
    __shared__ float gbuf[4][32][64];

    const int tid  = threadIdx.x;
    const int n0   = blockIdx.x * 64;
    const int wave = tid >> 5, lane = tid & 31;
    const int mt   = wave & 1;         // rows 0-15 or 16-31
    const int gate = wave >> 1;        // 0..3 (i,f,g,o)
    const int ml   = lane & 15;
    const int khalf = (lane >> 4) << 1;
    const int khp   = khalf >> 1;
    const int mhalf = (lane >> 4) << 3;

    const float* hprev = (t == 0) ? h0 : (hs + (size_t)(t - 1) * B_ * H_);

    const v8f vz = {0.f,0.f,0.f,0.f,0.f,0.f,0.f,0.f};
    v8f acc[4] = { vz, vz, vz, vz };

    const int lk = tid & 15;
    const int lr = tid >> 4;

    for (int kk = 0; kk < H_; kk += 16) {
        As[lr][lk]      = hprev[(size_t)lr * H_ + kk + lk];
        As[lr + 16][lk] = hprev[(size_t)(lr + 16) * H_ + kk + lk];
        #pragma unroll
        for (int i = 0; i < 16; ++i) {
            int c    = lr + i * 16;                        // 0..255
            int wrow = (c >> 6) * H_ + n0 + (c & 63);      // gate*H + n
            BsP[lk >> 1][c][lk & 1] = Whh[(size_t)wrow * H_ + kk + lk];
        }
        if (kk + 16 < H_)
            __builtin_prefetch(Whh + (size_t)((lr >> 6) * H_ + n0 + (lr & 63)) * H_ + kk + 16 + lk, 0, 1);
        __syncthreads();

        v2f af[4];
        v2f bf[4][4];
        #pragma unroll
        for (int k4 = 0; k4 < 4; ++k4) {
            int kb = k4 * 4 + khalf;
            int kp = k4 * 2 + khp;
            af[k4] = *reinterpret_cast<const v2f*>(&As[mt * 16 + ml][kb]);
            #pragma unroll
            for (int j = 0; j < 4; ++j)
                bf[k4][j] = *reinterpret_cast<const v2f*>(&BsP[kp][gate * 64 + j * 16 + ml][0]);
        }
        #pragma unroll
        for (int k4 = 0; k4 < 4; ++k4)
            #pragma unroll
            for (int j = 0; j < 4; ++j)
                acc[j] = __builtin_amdgcn_wmma_f32_16x16x4_f32(
                    false, af[k4], false, bf[k4][j], (short)0, acc[j], false, false);
        __syncthreads();
    }

    // exchange per-gate partials through LDS
    #pragma unroll
    for (int j = 0; j < 4; ++j)
        #pragma unroll
        for (int r = 0; r < 8; ++r)
            gbuf[gate][mt * 16 + r + mhalf][j * 16 + ml] = acc[j][r];
    __syncthreads();

    // cell update: 32 x 64 elements, 8 per thread
    #pragma unroll
    for (int i = 0; i < 8; ++i) {
        int idx = tid + i * 256;
        int b = idx >> 6, nl = idx & 63;
        int n = n0 + nl;
        size_t prow = ((size_t)t * B_ + b) * (size_t)G4_;
        float iv = gbuf[0][b][nl] + pre[prow + n];
        float fv = gbuf[1][b][nl] + pre[prow + H_ + n];
        float gv = gbuf[2][b][nl] + pre[prow + 2 * H_ + n];
        float ov = gbuf[3][b][nl] + pre[prow + 3 * H_ + n];
        float cp = (t == 0) ? c0[b * H_ + n] : cbuf[b * H_ + n];
        float cn = sigm_(fv) * cp + sigm_(iv) * tanhf(gv);
        float hv = sigm_(ov) * tanhf(cn);
        cbuf[b * H_ + n] = cn;
        hs[((size_t)t * B_ + b) * H_ + n] = hv;
    }
}

// ---------------------------------------------------------------------------
// In-place log-softmax over V=32000 per row; one block per row, online max+sum
// ---------------------------------------------------------------------------
__global__ __launch_bounds__(256) void logsoftmax_rows(float* __restrict__ out)
{
    __shared__ float sm[256], ss[256];
    const int tid = threadIdx.x;
    float* p = out + (size_t)blockIdx.x * V_;

    float lm = -3.0e38f, ls = 0.f;
    for (int i = tid; i < V_; i += 256) {
        float x  = p[i];
        float nm = fmaxf(lm, x);
        ls = ls * expf(lm - nm) + expf(x - nm);
        lm = nm;
    }
    sm[tid] = lm; ss[tid] = ls;
    __syncthreads();
    for (int off = 128; off > 0; off >>= 1) {
        if (tid < off) {
            float m2 = sm[tid + off], s2 = ss[tid + off];
            float nm = fmaxf(sm[tid], m2);
            ss[tid] = ss[tid] * expf(sm[tid] - nm) + s2 * expf(m2 - nm);
            sm[tid] = nm;
        }
        __syncthreads();
    }
    const float lse = sm[0] + logf(ss[0]);
    for (int i = tid; i < V_; i += 256) p[i] -= lse;
}

// ---------------------------------------------------------------------------
// Emit hn[2,B,H] and cn[2,B,H] into the output tail
// ---------------------------------------------------------------------------
__global__ __launch_bounds__(256) void finalize_states(
    const float* __restrict__ hs0, const float* __restrict__ hs1,
    const float* __restrict__ cb0, const float* __restrict__ cb1,
    float* __restrict__ out_tail)
{
    int idx = blockIdx.x * 256 + threadIdx.x;      // < 2*B*H = 65536
    int l   = idx >> 15;
    int rem = idx & 32767;                         // b*H + h
    const float* hsl = l ? hs1 : hs0;
    const float* cbl = l ? cb1 : cb0;
    out_tail[idx]         = hsl[(size_t)(T_ - 1) * B_ * H_ + rem];
    out_tail[65536 + idx] = cbl[rem];
}

// ---------------------------------------------------------------------------
extern "C" void kernel_launch(void* const* d_in, const int* in_sizes, int n_in,
                              void* d_out, int out_size, void* d_ws, size_t ws_size,
                              hipStream_t stream)
{
    const float* h0     = (const float*)d_in[1];   // [2,B,H]
    const float* c0     = (const float*)d_in[2];   // [2,B,H]
    const int*   labels = (const int*)  d_in[3];   // [B,T]
    const float* emb    = (const float*)d_in[4];   // [V,E]
    const float* W_ih0  = (const float*)d_in[5];
    const float* W_hh0  = (const float*)d_in[6];
    const float* b_ih0  = (const float*)d_in[7];
    const float* b_hh0  = (const float*)d_in[8];
    const float* W_ih1  = (const float*)d_in[9];
    const float* W_hh1  = (const float*)d_in[10];
    const float* b_ih1  = (const float*)d_in[11];
    const float* b_hh1  = (const float*)d_in[12];
    const float* W_out  = (const float*)d_in[13];
    const float* b_out  = (const float*)d_in[14];
    float* out = (float*)d_out;

    // workspace layout (floats): pre 16.7M | hs0 4.2M | hs1 4.2M | c0 32K | c1 32K
    float* ws  = (float*)d_ws;
    float* pre = ws;
    float* hs0 = pre + (size_t)T_ * B_ * G4_;
    float* hs1 = hs0 + (size_t)T_ * B_ * H_;
    float* cb0 = hs1 + (size_t)T_ * B_ * H_;
    float* cb1 = cb0 + (size_t)B_ * H_;

    dim3 blk(256);

    // layer 0: batched input contribution (embedding gather fused), then scan
    gemm_aw<<<dim3(64, 32), blk, 0, stream>>>(nullptr, 0, emb, labels,
        W_ih0, b_ih0, b_hh0, pre, (long)G4_, H_, 1, 0);
    for (int t = 0; t < T_; ++t)
        lstm_step<<<dim3(16), blk, 0, stream>>>(pre, W_hh0, h0, c0, hs0, cb0, t);

    // layer 1
    gemm_aw<<<dim3(64, 32), blk, 0, stream>>>(hs0, H_, nullptr, nullptr,
        W_ih1, b_ih1, b_hh1, pre, (long)G4_, H_, 0, 0);
    for (int t = 0; t < T_; ++t)
        lstm_step<<<dim3(16), blk, 0, stream>>>(pre, W_hh1,
            h0 + B_ * H_, c0 + B_ * H_, hs1, cb1, t);

    // projection to vocab (dominant GEMM), remapped [t,b] -> [b,t]
    gemm_aw<<<dim3(64, 250), blk, 0, stream>>>(hs1, H_, nullptr, nullptr,
        W_out, b_out, nullptr, out, (long)V_, H_, 0, 1);

    logsoftmax_rows<<<dim3(B_ * T_), blk, 0, stream>>>(out);

    finalize_states<<<dim3(256), blk, 0, stream>>>(hs0, hs1, cb0, cb1,
        out + (size_t)B_ * T_ * V_);
}